// CTCLoss_43215960932898
// MI455X (gfx1250) — compile-verified
//
#include <hip/hip_runtime.h>
#include <cstdint>

// CTC loss + focal reweighting for fixed shapes B=256, T=64, C=6625, S=25.
// predicts are treated directly as log-probs (same contract as the reference).

#define NEGV (-1e30f)

constexpr int kB  = 256;
constexpr int kT  = 64;
constexpr int kC  = 6625;
constexpr int kS  = 25;
constexpr int kS2 = 2 * kS + 1;   // 51 extended states
constexpr int kRow = 52;          // padded LDS row stride for lp[t][s]

// ---- CDNA5 async global -> LDS copy (per-lane gather), tracked by ASYNCcnt ----
// lds_off: raw LDS byte offset (low 32 bits of a flat pointer into __shared__).
__device__ __forceinline__ void async_ld_f32(const float* gaddr, uint32_t lds_off) {
  asm volatile("global_load_async_to_lds_b32 %0, %1, off"
               :: "v"(lds_off), "v"(gaddr) : "memory");
}
__device__ __forceinline__ void wait_async_all() {
  asm volatile("s_wait_asynccnt 0x0" ::: "memory");
}

// jnp.logaddexp-compatible: max + log1p(exp(min-max)); NEGV sentinel behaves
// like the reference (NEGV,NEGV -> NEGV + ln2 ~= NEGV).
__device__ __forceinline__ float lae(float x, float y) {
  float mx = fmaxf(x, y);
  float mn = fminf(x, y);
  return mx + log1pf(expf(mn - mx));
}

__global__ __launch_bounds__(64) void ctc_alpha_kernel(
    const float* __restrict__ predicts,   // [B,T,C] f32 log-probs
    const int*   __restrict__ labels,     // [B,S] i32 (1..C-1)
    const int*   __restrict__ lab_len,    // [B] i32
    float*       __restrict__ wloss)      // [B] focal-weighted loss out
{
  const int b   = blockIdx.x;
  const int tid = threadIdx.x;            // 64 threads = 2 wave32

  __shared__ int   ext[kS2];
  __shared__ float lp[kT * kRow];         // lp[t*kRow + s] = predicts[b,t,ext[s]]
  __shared__ float abuf[2][64];

  // Extended label sequence: blank, l1, blank, l2, ..., blank
  if (tid < kS2)
    ext[tid] = (tid & 1) ? labels[b * kS + (tid >> 1)] : 0;
  __syncthreads();

  // Async-gather the whole [T x S2] slab into LDS: thread s walks t with
  // stride C floats; 64 async b32 ops per wave, ASYNCcnt-tracked.
  if (tid < kS2) {
    const float* gp   = predicts + (size_t)b * kT * kC + ext[tid];
    uint32_t     lds0 = (uint32_t)(uintptr_t)(&lp[tid]);
#pragma unroll 4
    for (int t = 0; t < kT; ++t)
      async_ld_f32(gp + (size_t)t * kC, lds0 + (uint32_t)(t * kRow * 4));
  }

  // skip connection: ext[s]!=0 (odd s) and ext[s]!=ext[s-2], disabled for s<2
  bool skip = false;
  if ((tid & 1) && tid >= 3) skip = (ext[tid] != ext[tid - 2]);

  wait_async_all();      // wave-level wait for our async transfers
  __syncthreads();       // make LDS writes visible across both waves

  // alpha init at t=0
  abuf[0][tid] = (tid < 2) ? lp[tid] : NEGV;

  int cur = 0;
  for (int t = 1; t < kT; ++t) {
    __syncthreads();     // separates prev write / this read (double-buffered)
    float an = NEGV;
    if (tid < kS2) {
      float a0 = abuf[cur][tid];
      float a1 = (tid >= 1) ? abuf[cur][tid - 1] : NEGV;
      float a2 = skip ? abuf[cur][tid - 2] : NEGV;
      an = lae(lae(a0, a1), a2) + lp[t * kRow + tid];
    }
    cur ^= 1;
    abuf[cur][tid] = an;
  }
  __syncthreads();

  if (tid == 0) {
    int L  = lab_len[b];            // 1..25
    int i1 = 2 * L;                 // <= 50
    int i0 = i1 - 1; if (i0 < 0) i0 = 0;
    float loss = -lae(abuf[cur][i1], abuf[cur][i0]);
    float w    = 1.0f - expf(-loss);          // focal weight (1-e^-loss)^2
    wloss[b]   = loss * w * w;
  }
}

// ---- mean over B via one wave32; 32->16 partial fold done with a real WMMA ----
typedef __attribute__((ext_vector_type(2))) float v2f;
typedef __attribute__((ext_vector_type(8))) float v8f;

__global__ __launch_bounds__(32) void focal_mean_kernel(const float* __restrict__ wloss,
                                                        float* __restrict__ out) {
  const int lane = threadIdx.x;     // exactly one wave, EXEC all-ones for WMMA
  float p = 0.f;
#pragma unroll
  for (int i = 0; i < 8; ++i) p += wloss[lane * 8 + i];

  // D = A(16x4, all ones) x B(4x16, partials) + 0  => every D row = column
  // sums of B; summing D[0][0..15] recovers the exact total regardless of the
  // precise lane->(k,n) slot mapping, since A is all-ones and unused B slots
  // are zero. f32 WMMA keeps full precision.
  v2f a;  a.x = 1.0f; a.y = 1.0f;
  v2f bm; bm.x = p;   bm.y = 0.0f;
  v8f c = {};
  c = __builtin_amdgcn_wmma_f32_16x16x4_f32(false, a, false, bm,
                                            (short)0, c, false, false);
  float v = c[0];                   // lane n (0..15): D[0][n]
  v += __shfl_down(v, 8);
  v += __shfl_down(v, 4);
  v += __shfl_down(v, 2);
  v += __shfl_down(v, 1);
  if (lane == 0) out[0] = v * (1.0f / (float)kB);
}

extern "C" void kernel_launch(void* const* d_in, const int* in_sizes, int n_in,
                              void* d_out, int out_size, void* d_ws, size_t ws_size,
                              hipStream_t stream) {
  const float* predicts = (const float*)d_in[0];
  const int*   labels   = (const int*)d_in[1];
  const int*   lab_len  = (const int*)d_in[2];   // int32 view (values 1..25)
  float*       wloss    = (float*)d_ws;          // 256 floats of scratch

  ctc_alpha_kernel<<<kB, 64, 0, stream>>>(predicts, labels, lab_len, wloss);
  focal_mean_kernel<<<1, 32, 0, stream>>>(wloss, (float*)d_out);
}